// LightGCNLayer_7868380086590
// MI455X (gfx1250) — compile-verified
//
#include <hip/hip_runtime.h>
#include <stdint.h>

#define D_FEAT   128     // feature dim -> 32 lanes x float4 per row
#define EPB_FILL 256     // edges per block in the CSR fill pass
#define SCAN_T   1024    // threads in the single-block prefix-sum kernel

// ---------------------------------------------------------------------------
// Kernel 1: zero the two counter arrays (deg over rows, cnt over cols).
// Everything else (offsets/cursor/src/norm/out) is fully overwritten each call.
// ---------------------------------------------------------------------------
__global__ void lightgcn_zero_cnt(uint32_t* __restrict__ deg_row,
                                  uint32_t* __restrict__ cnt_col,
                                  int num_nodes) {
  int i      = blockIdx.x * blockDim.x + threadIdx.x;
  int stride = gridDim.x * blockDim.x;
  for (int k = i; k < num_nodes; k += stride) {
    deg_row[k] = 0u;
    cnt_col[k] = 0u;
  }
}

// ---------------------------------------------------------------------------
// Kernel 2: count out-degree (rows) and in-count (cols). 2 u32 atomics/edge.
// ---------------------------------------------------------------------------
__global__ void lightgcn_count(const int* __restrict__ rows,
                               const int* __restrict__ cols,
                               int num_edges,
                               uint32_t* __restrict__ deg_row,
                               uint32_t* __restrict__ cnt_col) {
  int e = blockIdx.x * blockDim.x + threadIdx.x;
  if (e < num_edges) {
    __hip_atomic_fetch_add(&deg_row[rows[e]], 1u, __ATOMIC_RELAXED,
                           __HIP_MEMORY_SCOPE_AGENT);   // global_atomic_add_u32 (no-ret)
    __hip_atomic_fetch_add(&cnt_col[cols[e]], 1u, __ATOMIC_RELAXED,
                           __HIP_MEMORY_SCOPE_AGENT);
  }
}

// ---------------------------------------------------------------------------
// Kernel 3: dinv[v] = 1/sqrt(max(deg_row[v],1))
// ---------------------------------------------------------------------------
__global__ void lightgcn_dinv(const uint32_t* __restrict__ deg,
                              float* __restrict__ dinv, int num_nodes) {
  int v = blockIdx.x * blockDim.x + threadIdx.x;
  if (v < num_nodes) {
    float d = (float)deg[v];
    d = (d < 1.0f) ? 1.0f : d;
    dinv[v] = 1.0f / sqrtf(d);
  }
}

// ---------------------------------------------------------------------------
// Kernel 4: single-block exclusive prefix sum over cnt_col -> offsets, cursor.
//   1024 threads, each owns a contiguous chunk (~98 elems): serial local sum,
//   LDS Hillis-Steele scan of the 1024 totals, then serial local rescan.
// ---------------------------------------------------------------------------
__global__ void __launch_bounds__(SCAN_T)
lightgcn_scan(const uint32_t* __restrict__ cnt,
              uint32_t* __restrict__ offsets,   // [n+1]
              uint32_t* __restrict__ cursor,    // [n]
              int n) {
  __shared__ uint32_t tot[SCAN_T];
  const int t     = threadIdx.x;
  const int chunk = (n + SCAN_T - 1) / SCAN_T;
  const int b     = t * chunk;
  const int e     = (b + chunk < n) ? (b + chunk) : n;

  uint32_t s = 0;
  for (int i = b; i < e; ++i) s += cnt[i];
  tot[t] = s;
  __syncthreads();

  // inclusive scan of per-thread totals
  for (int off = 1; off < SCAN_T; off <<= 1) {
    uint32_t v = (t >= off) ? tot[t - off] : 0u;
    __syncthreads();
    tot[t] += v;
    __syncthreads();
  }

  uint32_t run = tot[t] - s;          // exclusive prefix of this chunk
  for (int i = b; i < e; ++i) {
    uint32_t cv = cnt[i];
    offsets[i] = run;
    cursor[i]  = run;
    run += cv;
  }
  if (t == SCAN_T - 1) offsets[n] = run;   // total edge count
}

// ---------------------------------------------------------------------------
// Kernel 5: CSR fill. Block stages its 256 edges' (row,col) indices into LDS
// via GLOBAL_LOAD_ASYNC_TO_LDS_B32 (ASYNCcnt), then each thread places its
// edge: p = cursor[col]++ (returning u32 atomic); src[p]=row; norm[p]=
// dinv[row]*dinv[col]. Moves all random dinv reads into this streamed pass.
// ---------------------------------------------------------------------------
__global__ void __launch_bounds__(EPB_FILL)
lightgcn_fill(const int* __restrict__ rows,
              const int* __restrict__ cols,
              const float* __restrict__ dinv,
              uint32_t* __restrict__ cursor,
              int* __restrict__ src_sorted,
              float* __restrict__ norm_sorted,
              int num_edges) {
  __shared__ int sidx[2 * EPB_FILL];   // [0..EPB): rows, [EPB..2*EPB): cols

  const int t  = threadIdx.x;
  const int e0 = blockIdx.x * EPB_FILL;

  {
    int e = e0 + t;
    if (e >= num_edges) e = num_edges - 1;   // clamp: legal addr for tail lanes
#if defined(__gfx1250__)
    uint32_t lds_r = (uint32_t)(uintptr_t)&sidx[t];
    uint32_t lds_c = (uint32_t)(uintptr_t)&sidx[EPB_FILL + t];
    uint64_t ga_r  = (uint64_t)(uintptr_t)(rows + e);
    uint64_t ga_c  = (uint64_t)(uintptr_t)(cols + e);
    asm volatile("global_load_async_to_lds_b32 %0, %1, off"
                 :: "v"(lds_r), "v"(ga_r) : "memory");
    asm volatile("global_load_async_to_lds_b32 %0, %1, off"
                 :: "v"(lds_c), "v"(ga_c) : "memory");
#else
    sidx[t]            = rows[e];
    sidx[EPB_FILL + t] = cols[e];
#endif
  }
#if defined(__gfx1250__)
  asm volatile("s_wait_asynccnt 0" ::: "memory");
#endif
  __syncthreads();

  const int e = e0 + t;
  if (e >= num_edges) return;

  const int r = sidx[t];
  const int c = sidx[EPB_FILL + t];
  const float w = dinv[r] * dinv[c];

  uint32_t p = __hip_atomic_fetch_add(&cursor[c], 1u, __ATOMIC_RELAXED,
                                      __HIP_MEMORY_SCOPE_AGENT);  // returning atomic
  src_sorted[p]  = r;
  norm_sorted[p] = w;
}

// ---------------------------------------------------------------------------
// Kernel 6: gather. One wave32 per destination node; lane l owns floats
// [4l,4l+4). Register accumulation with fmaf, zero atomics, one b128 store.
// Lane 0 prefetches the source row two CSR entries ahead (global_prefetch_b8).
// ---------------------------------------------------------------------------
__global__ void __launch_bounds__(256)
lightgcn_gather(const float* __restrict__ emb,
                const uint32_t* __restrict__ offsets,
                const int* __restrict__ src_sorted,
                const float* __restrict__ norm_sorted,
                float* __restrict__ out,
                int num_nodes) {
  int tid  = blockIdx.x * blockDim.x + threadIdx.x;
  int v    = tid >> 5;
  int lane = tid & 31;
  if (v >= num_nodes) return;

  const uint32_t beg = offsets[v];
  const uint32_t end = offsets[v + 1];

  float4 acc = make_float4(0.0f, 0.0f, 0.0f, 0.0f);
  for (uint32_t j = beg; j < end; ++j) {
    if (lane == 0 && (j + 2) < end) {
      __builtin_prefetch(&emb[(size_t)src_sorted[j + 2] * D_FEAT], 0, 1); // global_prefetch_b8
    }
    const int   r = src_sorted[j];
    const float w = norm_sorted[j];
    const float4 x =
        ((const float4*)(emb + (size_t)r * D_FEAT))[lane];        // global_load_b128
    acc.x = fmaf(x.x, w, acc.x);
    acc.y = fmaf(x.y, w, acc.y);
    acc.z = fmaf(x.z, w, acc.z);
    acc.w = fmaf(x.w, w, acc.w);
  }
  ((float4*)(out + (size_t)v * D_FEAT))[lane] = acc;              // global_store_b128
}

// ---------------------------------------------------------------------------
// Launcher
//   d_in[0]: embeddings, float32, [N, 128]     (in_sizes[0] = N*128)
//   d_in[1]: edge_index, int32,   [2, E]       (in_sizes[1] = 2*E; row-major)
//   d_out  : float32 [N, 128]
//   d_ws layout (4-byte units):
//     dinv[N] | deg_row[N] | cnt_col[N] | offsets[N+1] | cursor[N]
//     | src_sorted[E] | norm_sorted[E]           (~7.2 MB total)
// ---------------------------------------------------------------------------
extern "C" void kernel_launch(void* const* d_in, const int* in_sizes, int n_in,
                              void* d_out, int out_size, void* d_ws, size_t ws_size,
                              hipStream_t stream) {
  const float* emb  = (const float*)d_in[0];
  const int*   eidx = (const int*)d_in[1];
  const int num_nodes = in_sizes[0] / D_FEAT;
  const int num_edges = in_sizes[1] / 2;

  const int* rows = eidx;              // edge_index[0, :]
  const int* cols = eidx + num_edges;  // edge_index[1, :]

  float* out = (float*)d_out;

  uint32_t* ws = (uint32_t*)d_ws;
  float*    dinv        = (float*)   (ws);
  uint32_t* deg_row     = (uint32_t*)(ws + (size_t)num_nodes);
  uint32_t* cnt_col     = (uint32_t*)(ws + (size_t)2 * num_nodes);
  uint32_t* offsets     = (uint32_t*)(ws + (size_t)3 * num_nodes);       // N+1
  uint32_t* cursor      = (uint32_t*)(ws + (size_t)4 * num_nodes + 1);
  int*      src_sorted  = (int*)     (ws + (size_t)5 * num_nodes + 2);
  float*    norm_sorted = (float*)   (ws + (size_t)5 * num_nodes + 2 + num_edges);

  lightgcn_zero_cnt<<<256, 256, 0, stream>>>(deg_row, cnt_col, num_nodes);
  lightgcn_count<<<(num_edges + 255) / 256, 256, 0, stream>>>(
      rows, cols, num_edges, deg_row, cnt_col);
  lightgcn_dinv<<<(num_nodes + 255) / 256, 256, 0, stream>>>(
      deg_row, dinv, num_nodes);
  lightgcn_scan<<<1, SCAN_T, 0, stream>>>(cnt_col, offsets, cursor, num_nodes);
  lightgcn_fill<<<(num_edges + EPB_FILL - 1) / EPB_FILL, EPB_FILL, 0, stream>>>(
      rows, cols, dinv, cursor, src_sorted, norm_sorted, num_edges);

  const long gather_threads = (long)num_nodes * 32;
  lightgcn_gather<<<(int)((gather_threads + 255) / 256), 256, 0, stream>>>(
      emb, offsets, src_sorted, norm_sorted, out, num_nodes);
}